// EdgeNetDeeper_7456063226143
// MI455X (gfx1250) — compile-verified
//
#include <hip/hip_runtime.h>

typedef __attribute__((ext_vector_type(2))) float v2f;
typedef __attribute__((ext_vector_type(8))) float v8f;

__device__ __forceinline__ v8f wmma_f32(v2f a, v2f b, v8f c) {
  // D = A(16x4) * B(4x16) + C(16x16), fp32, wave32
  return __builtin_amdgcn_wmma_f32_16x16x4_f32(false, a, false, b, (short)0, c, false, false);
}

// ---------------- BatchNorm: stats reduction ----------------
__global__ __launch_bounds__(256) void bn_reduce_kernel(const float* __restrict__ x,
                                                        float* __restrict__ stats, int N) {
  __shared__ float red[256][8];
  const int tid = threadIdx.x;
  const int n = blockIdx.x * 256 + tid;
  float v0 = 0.f, v1 = 0.f, v2 = 0.f, v3 = 0.f;
  if (n < N) {
    float4 xv = ((const float4*)x)[n];
    v0 = xv.x; v1 = xv.y; v2 = xv.z; v3 = xv.w;
  }
  red[tid][0] = v0; red[tid][1] = v1; red[tid][2] = v2; red[tid][3] = v3;
  red[tid][4] = v0 * v0; red[tid][5] = v1 * v1; red[tid][6] = v2 * v2; red[tid][7] = v3 * v3;
  __syncthreads();
  for (int s = 128; s > 0; s >>= 1) {
    if (tid < s) {
#pragma unroll
      for (int c = 0; c < 8; ++c) red[tid][c] += red[tid + s][c];
    }
    __syncthreads();
  }
  if (tid < 8) atomicAdd(&stats[tid], red[0][tid]);
}

// ---------------- BatchNorm: apply ----------------
__global__ __launch_bounds__(256) void bn_apply_kernel(const float* __restrict__ x,
                                                       const float* __restrict__ stats,
                                                       const float* __restrict__ gamma,
                                                       const float* __restrict__ beta,
                                                       float* __restrict__ h, int N) {
  const int idx = blockIdx.x * 256 + threadIdx.x;
  if (idx >= N * 4) return;
  const int c = idx & 3;
  const float invN = 1.0f / (float)N;
  const float mu = stats[c] * invN;
  const float var = stats[4 + c] * invN - mu * mu;
  h[idx] = (x[idx] - mu) * rsqrtf(var + 1e-5f) * gamma[c] + beta[c];
}

// ---------------- degree (same for all layers) ----------------
__global__ __launch_bounds__(256) void degree_kernel(const int* __restrict__ dst,
                                                     float* __restrict__ deg, int E) {
  const int e = blockIdx.x * 256 + threadIdx.x;
  if (e < E) atomicAdd(&deg[dst[e]], 1.0f);
}

// ---------------- finalize: scatter-mean divide ----------------
template <int D>
__global__ __launch_bounds__(256) void finalize_kernel(const float* __restrict__ acc,
                                                       const float* __restrict__ deg,
                                                       float* __restrict__ out, int N) {
  const int idx = blockIdx.x * 256 + threadIdx.x;
  if (idx >= N * D) return;
  out[idx] = acc[idx] / fmaxf(deg[idx / D], 1.0f);
}

// ---------------- EdgeConv: 16-edge tile per wave, WMMA f32 MLP ----------------
// msg = MLP([x_i, x_j - x_i]); acc[dst] += msg (atomics)
template <int K_IN, int F, int DOUT, bool RELU>
__global__ __launch_bounds__(128) void edgeconv_kernel(
    const float* __restrict__ h, const int* __restrict__ src, const int* __restrict__ dst,
    const float* __restrict__ W1, const float* __restrict__ B1,
    const float* __restrict__ W2, const float* __restrict__ B2,
    const float* __restrict__ W3, const float* __restrict__ B3,
    float* __restrict__ acc, int E) {
  constexpr int NW = 4;                        // waves per block
  constexpr int NT3 = (DOUT > 16) ? 2 : 1;     // n-tiles for layer 3
  constexpr int W3N = 16 * NT3;                // padded layer-3 width
  constexpr int AROW = (K_IN > 32) ? K_IN : 32;

  __shared__ float sW1[K_IN * 32];
  __shared__ float sW2[32 * 32];
  __shared__ float sW3[32 * W3N];
  __shared__ float sB1[32];
  __shared__ float sB2[32];
  __shared__ float sB3[W3N];
  __shared__ float bufA[NW][16 * AROW];
  __shared__ float bufB[NW][16 * 32];
  __shared__ int sDst[NW][16];

  const int tid = threadIdx.x;
  for (int i = tid; i < K_IN * 32; i += 128) sW1[i] = W1[i];
  for (int i = tid; i < 32 * 32; i += 128) sW2[i] = W2[i];
  for (int i = tid; i < 32 * W3N; i += 128) {
    int k = i / W3N, n = i % W3N;
    sW3[i] = (n < DOUT) ? W3[k * DOUT + n] : 0.0f;
  }
  if (tid < 32) { sB1[tid] = B1[tid]; sB2[tid] = B2[tid]; }
  if (tid < W3N) sB3[tid] = (tid < DOUT) ? B3[tid] : 0.0f;
  __syncthreads();

  const int wave = tid >> 5;
  const int lane = tid & 31;
  const int half = lane >> 4;  // 0: K pair 0-1, 1: K pair 2-3 (A/B operand layout)
  const int mrow = lane & 15;
  const int col0 = mrow;
  const int col1 = 16 + mrow;

  const int tile = blockIdx.x * NW + wave;
  const int e0 = tile * 16;
  if (e0 >= E) return;  // wave-uniform: EXEC stays all-ones for WMMA

  if (lane < 16) {
    const int e = e0 + lane;
    sDst[wave][lane] = (e < E) ? dst[e] : -1;
  }

  {  // stage input tile 16 x K_IN = [x_i | x_j - x_i]
    const int e = e0 + mrow;
    const bool valid = (e < E);
    const int vi = valid ? dst[e] : 0;
    float* row = &bufA[wave][mrow * K_IN];
    if (half == 0) {
      if constexpr ((F % 4) == 0) {
        const float4* hv = (const float4*)h;
#pragma unroll
        for (int q = 0; q < F / 4; ++q) {
          float4 z = make_float4(0.f, 0.f, 0.f, 0.f);
          float4 xi = valid ? hv[vi * (F / 4) + q] : z;
          row[q * 4 + 0] = xi.x; row[q * 4 + 1] = xi.y;
          row[q * 4 + 2] = xi.z; row[q * 4 + 3] = xi.w;
        }
      } else {
#pragma unroll
        for (int f = 0; f < F; ++f) row[f] = valid ? h[vi * F + f] : 0.0f;
      }
    } else {
      const int vj = valid ? src[e] : 0;
      if constexpr ((F % 4) == 0) {
        const float4* hv = (const float4*)h;
#pragma unroll
        for (int q = 0; q < F / 4; ++q) {
          float4 z = make_float4(0.f, 0.f, 0.f, 0.f);
          float4 xi = valid ? hv[vi * (F / 4) + q] : z;
          float4 xj = valid ? hv[vj * (F / 4) + q] : z;
          row[F + q * 4 + 0] = xj.x - xi.x; row[F + q * 4 + 1] = xj.y - xi.y;
          row[F + q * 4 + 2] = xj.z - xi.z; row[F + q * 4 + 3] = xj.w - xi.w;
        }
      } else {
#pragma unroll
        for (int f = 0; f < F; ++f) {
          float xi = valid ? h[vi * F + f] : 0.0f;
          float xj = valid ? h[vj * F + f] : 0.0f;
          row[F + f] = xj - xi;
        }
      }
    }
  }

  // ---- layer 1: K_IN -> 32, ReLU
  v8f c0 = {0, 0, 0, 0, 0, 0, 0, 0};
  v8f c1 = {0, 0, 0, 0, 0, 0, 0, 0};
#pragma unroll
  for (int k0 = 0; k0 < K_IN; k0 += 4) {
    const int ka = k0 + 2 * half;
    v2f a; a.x = bufA[wave][mrow * K_IN + ka]; a.y = bufA[wave][mrow * K_IN + ka + 1];
    v2f b0; b0.x = sW1[ka * 32 + col0]; b0.y = sW1[(ka + 1) * 32 + col0];
    v2f b1; b1.x = sW1[ka * 32 + col1]; b1.y = sW1[(ka + 1) * 32 + col1];
    c0 = wmma_f32(a, b0, c0);
    c1 = wmma_f32(a, b1, c1);
  }
#pragma unroll
  for (int r = 0; r < 8; ++r) {
    const int m = r + 8 * half;
    bufB[wave][m * 32 + col0] = fmaxf(c0[r] + sB1[col0], 0.0f);
    bufB[wave][m * 32 + col1] = fmaxf(c1[r] + sB1[col1], 0.0f);
  }

  // ---- layer 2: 32 -> 32, ReLU
  v8f d0 = {0, 0, 0, 0, 0, 0, 0, 0};
  v8f d1 = {0, 0, 0, 0, 0, 0, 0, 0};
#pragma unroll
  for (int k0 = 0; k0 < 32; k0 += 4) {
    const int ka = k0 + 2 * half;
    v2f a; a.x = bufB[wave][mrow * 32 + ka]; a.y = bufB[wave][mrow * 32 + ka + 1];
    v2f b0; b0.x = sW2[ka * 32 + col0]; b0.y = sW2[(ka + 1) * 32 + col0];
    v2f b1; b1.x = sW2[ka * 32 + col1]; b1.y = sW2[(ka + 1) * 32 + col1];
    d0 = wmma_f32(a, b0, d0);
    d1 = wmma_f32(a, b1, d1);
  }
#pragma unroll
  for (int r = 0; r < 8; ++r) {
    const int m = r + 8 * half;
    bufA[wave][m * 32 + col0] = fmaxf(d0[r] + sB2[col0], 0.0f);
    bufA[wave][m * 32 + col1] = fmaxf(d1[r] + sB2[col1], 0.0f);
  }

  // ---- layer 3: 32 -> DOUT (padded to W3N), optional ReLU
  v8f g0 = {0, 0, 0, 0, 0, 0, 0, 0};
  v8f g1 = {0, 0, 0, 0, 0, 0, 0, 0};
#pragma unroll
  for (int k0 = 0; k0 < 32; k0 += 4) {
    const int ka = k0 + 2 * half;
    v2f a; a.x = bufA[wave][mrow * 32 + ka]; a.y = bufA[wave][mrow * 32 + ka + 1];
    v2f b0; b0.x = sW3[ka * W3N + col0]; b0.y = sW3[(ka + 1) * W3N + col0];
    g0 = wmma_f32(a, b0, g0);
    if constexpr (NT3 == 2) {
      v2f b1; b1.x = sW3[ka * W3N + col1]; b1.y = sW3[(ka + 1) * W3N + col1];
      g1 = wmma_f32(a, b1, g1);
    }
  }

  // ---- scatter-add messages into accumulator
#pragma unroll
  for (int r = 0; r < 8; ++r) {
    const int m = r + 8 * half;
    const int node = sDst[wave][m];
    if (node < 0) continue;
    float v0 = g0[r] + sB3[col0];
    if (RELU) v0 = fmaxf(v0, 0.0f);
    if constexpr (NT3 == 2) {
      float v1 = g1[r] + sB3[col1];
      if (RELU) v1 = fmaxf(v1, 0.0f);
      atomicAdd(&acc[node * DOUT + col0], v0);
      atomicAdd(&acc[node * DOUT + col1], v1);
    } else {
      if (col0 < DOUT) atomicAdd(&acc[node * DOUT + col0], v0);
    }
  }
}

extern "C" void kernel_launch(void* const* d_in, const int* in_sizes, int n_in,
                              void* d_out, int out_size, void* d_ws, size_t ws_size,
                              hipStream_t stream) {
  const float* x = (const float*)d_in[0];
  const int* edge_index = (const int*)d_in[1];
  const float* gamma = (const float*)d_in[2];
  const float* beta = (const float*)d_in[3];
  const float* P[24];
  for (int i = 0; i < 24; ++i) P[i] = (const float*)d_in[4 + i];

  const int N = in_sizes[0] / 4;
  const int E = in_sizes[1] / 2;
  const int* src = edge_index;       // row 0 = source j
  const int* dstp = edge_index + E;  // row 1 = target i

  // workspace layout (floats): [stats 8 (pad 64)][deg N][hA N*32][hB N*32][acc N*32]
  float* ws = (float*)d_ws;
  float* stats = ws;
  float* deg = ws + 64;
  float* hA = deg + N;
  float* hB = hA + (size_t)N * 32;
  float* accb = hB + (size_t)N * 32;

  hipMemsetAsync(ws, 0, (size_t)(64 + N) * sizeof(float), stream);

  bn_reduce_kernel<<<(N + 255) / 256, 256, 0, stream>>>(x, stats, N);
  bn_apply_kernel<<<(N * 4 + 255) / 256, 256, 0, stream>>>(x, stats, gamma, beta, hA, N);
  degree_kernel<<<(E + 255) / 256, 256, 0, stream>>>(dstp, deg, E);

  const int tiles = (E + 15) / 16;
  const int egrid = (tiles + 3) / 4;  // NW = 4 waves/block, 128 threads

  // enc1: F=4 (K=8) -> 32, final ReLU
  hipMemsetAsync(accb, 0, (size_t)N * 32 * sizeof(float), stream);
  edgeconv_kernel<8, 4, 32, true><<<egrid, 128, 0, stream>>>(
      hA, src, dstp, P[0], P[1], P[2], P[3], P[4], P[5], accb, E);
  finalize_kernel<32><<<(N * 32 + 255) / 256, 256, 0, stream>>>(accb, deg, hB, N);

  // enc2: F=32 (K=64) -> 2, final ReLU
  hipMemsetAsync(accb, 0, (size_t)N * 2 * sizeof(float), stream);
  edgeconv_kernel<64, 32, 2, true><<<egrid, 128, 0, stream>>>(
      hB, src, dstp, P[6], P[7], P[8], P[9], P[10], P[11], accb, E);
  finalize_kernel<2><<<(N * 2 + 255) / 256, 256, 0, stream>>>(accb, deg, hA, N);

  // dec1: F=2 (K=4) -> 32, final ReLU
  hipMemsetAsync(accb, 0, (size_t)N * 32 * sizeof(float), stream);
  edgeconv_kernel<4, 2, 32, true><<<egrid, 128, 0, stream>>>(
      hA, src, dstp, P[12], P[13], P[14], P[15], P[16], P[17], accb, E);
  finalize_kernel<32><<<(N * 32 + 255) / 256, 256, 0, stream>>>(accb, deg, hB, N);

  // dec2: F=32 (K=64) -> 4, no final ReLU
  hipMemsetAsync(accb, 0, (size_t)N * 4 * sizeof(float), stream);
  edgeconv_kernel<64, 32, 4, false><<<egrid, 128, 0, stream>>>(
      hB, src, dstp, P[18], P[19], P[20], P[21], P[22], P[23], accb, E);
  finalize_kernel<4><<<(N * 4 + 255) / 256, 256, 0, stream>>>(accb, deg, (float*)d_out, N);
}